// BPDecoder_31731218383095
// MI455X (gfx1250) — compile-verified
//
#include <hip/hip_runtime.h>
#include <math.h>

// ---- CDNA5 (gfx1250, wave32) vector types ----
typedef __attribute__((ext_vector_type(2))) float v2f;
typedef __attribute__((ext_vector_type(4))) float v4f;
typedef __attribute__((ext_vector_type(8))) float v8f;

#define NB  512   // blocks in pass 1 (needs NB*4 bytes of d_ws)
#define TPB 256   // 8 wave32 waves per block
#define WPB (TPB / 32)

// Reduce each wave's 32 fp32 partials with one V_WMMA_F32_16X16X4_F32:
//   A(16x4): lane l<16 holds A[l][0]=p, A[l][1]=0 ; lane l>=16 holds A[l-16][2]=p, A[l-16][3]=0
//   B(4x16): all ones  =>  D[m][n] = p_m + p_{m+16}  (same for every column n)
// C/D layout: lanes 0..15 hold columns N=lane, VGPR r -> M=r; lanes 16..31 -> M=r+8.
// So (sum of lane0's 8 regs) + (sum of lane16's 8 regs) = full wave total, exactly in fp32.
__device__ __forceinline__ float wave_partial_colsum(float p) {
    v2f a; a.x = p;   a.y = 0.0f;
    v2f b; b.x = 1.0f; b.y = 1.0f;
    v8f c = {};
    c = __builtin_amdgcn_wmma_f32_16x16x4_f32(
            /*neg_a=*/false, a, /*neg_b=*/false, b,
            /*c_mod=*/(short)0, c, /*reuse_a=*/false, /*reuse_b=*/false);
    float t = c[0] + c[1];
    t += c[2] + c[3];
    t += c[4] + c[5];
    t += c[6] + c[7];
    return t;  // lanes 0 and 16 together hold the wave total
}

__device__ __forceinline__ float row_term(const v4f* __restrict__ Zv,
                                          const float* __restrict__ R,
                                          int row, int lane) {
    const int srow = __builtin_amdgcn_readfirstlane(row);            // scalarize
    const v4f z = __builtin_nontemporal_load(&Zv[srow * 32 + lane]); // b128, TH=NT
    const float r = R[srow];
    float d = z.x * z.x;
    d = fmaf(z.y, z.y, d);
    d = fmaf(z.z, z.z, d);
    d = fmaf(z.w, z.w, d);
    return r * d;
}

__global__ void __launch_bounds__(TPB)
bp_rownorm_pass1(const float* __restrict__ Z, const float* __restrict__ R,
                 float* __restrict__ partials, int n) {
    const int lane   = threadIdx.x & 31;
    const int wid    = threadIdx.x >> 5;
    const int gwave  = blockIdx.x * WPB + wid;
    const int nwaves = gridDim.x * WPB;

    // One wave == one row: 32 lanes x float4 = 128 floats (D), coalesced b128.
    const v4f* __restrict__ Zv = (const v4f*)Z;
    float p = 0.0f;
    int row = gwave;

    // 4-row unroll: 4 independent b128 loads in flight per wave (MLP for the
    // 23.3 TB/s streaming floor) before any s_wait_loadcnt.
    for (; row + 3 * nwaves < n; row += 4 * nwaves) {
        const float t0 = row_term(Zv, R, row,              lane);
        const float t1 = row_term(Zv, R, row +     nwaves, lane);
        const float t2 = row_term(Zv, R, row + 2 * nwaves, lane);
        const float t3 = row_term(Zv, R, row + 3 * nwaves, lane);
        p += (t0 + t1) + (t2 + t3);
    }
    for (; row < n; row += nwaves) {
        p += row_term(Zv, R, row, lane);
    }

    // Wave reduction via fp32 WMMA (EXEC all-ones: loop bounds are wave-uniform).
    const float t = wave_partial_colsum(p);

    __shared__ float sm[2 * WPB];
    if (lane == 0 || lane == 16) sm[wid * 2 + (lane >> 4)] = t;
    __syncthreads();
    if (threadIdx.x == 0) {
        float s = 0.0f;
        #pragma unroll
        for (int i = 0; i < 2 * WPB; ++i) s += sm[i];
        partials[blockIdx.x] = s;
    }
}

__global__ void __launch_bounds__(TPB)
bp_rownorm_pass2(const float* __restrict__ partials, float* __restrict__ out) {
    const int lane = threadIdx.x & 31;
    const int wid  = threadIdx.x >> 5;

    float p = 0.0f;
    #pragma unroll
    for (int i = threadIdx.x; i < NB; i += TPB) p += partials[i]; // exactly 2 each, no divergence

    const float t = wave_partial_colsum(p);

    __shared__ float sm[2 * WPB];
    if (lane == 0 || lane == 16) sm[wid * 2 + (lane >> 4)] = t;
    __syncthreads();
    if (threadIdx.x == 0) {
        float s = 0.0f;
        #pragma unroll
        for (int i = 0; i < 2 * WPB; ++i) s += sm[i];
        const float lambdaterm = expf(s);
        out[0] = 1.0f - expf(-lambdaterm);
    }
}

extern "C" void kernel_launch(void* const* d_in, const int* in_sizes, int n_in,
                              void* d_out, int out_size, void* d_ws, size_t ws_size,
                              hipStream_t stream) {
    const float* Z = (const float*)d_in[0];   // (N, 128) fp32
    const float* R = (const float*)d_in[1];   // (N,)     fp32
    float* out      = (float*)d_out;          // scalar fp32
    float* partials = (float*)d_ws;           // NB floats (2 KB)
    const int n = in_sizes[1];                // N from R's element count

    bp_rownorm_pass1<<<NB, TPB, 0, stream>>>(Z, R, partials, n);
    bp_rownorm_pass2<<<1, TPB, 0, stream>>>(partials, out);
}